// SparseLayer_34076270526745
// MI455X (gfx1250) — compile-verified
//
#include <hip/hip_runtime.h>

// ---------------- problem constants ----------------
#define INPUT_SIZE 1024
#define OUTPUT_SIZE 1024
#define N_GAUSS 1024
#define N_LOCAL 8
#define N_GLOBAL 8
#define BATCH 4096
#define P_NEAR (4 * N_GAUSS)
#define P_LOC (N_LOCAL * N_GAUSS)
#define P_GLOB (N_GLOBAL * N_GAUSS)
#define NPTS (P_NEAR + P_LOC + P_GLOB) /* 20480 */
#define TAU 0.1f
#define SIGMA_BOOST 2.0f

typedef __attribute__((ext_vector_type(2))) float v2f;
typedef __attribute__((ext_vector_type(4))) float v4f;
typedef __attribute__((ext_vector_type(8))) float v8f;
typedef __attribute__((ext_vector_type(4))) int v4i;

// CDNA5 async global->LDS copies (ASYNCcnt-tracked), guarded so a missing
// builtin degrades to the synchronous path instead of breaking the build.
#if defined(__has_builtin)
#if __has_builtin(__builtin_amdgcn_global_load_async_to_lds_b128)
#define HAVE_ASYNC_LDS 1
#endif
#if __has_builtin(__builtin_amdgcn_s_wait_asynccnt)
#define HAVE_WAIT_ASYNC 1
#endif
#endif

#ifdef HAVE_ASYNC_LDS
// builtin signature (from hipcc diagnostic): first arg is v4i* in the global
// address space, second is v4i* in LDS, then imm offset + imm cpol.
typedef v4i __attribute__((address_space(1))) GV4i;
typedef v4i __attribute__((address_space(3))) LV4i;
#endif

__device__ __forceinline__ void wait_async_zero() {
#ifdef HAVE_WAIT_ASYNC
  __builtin_amdgcn_s_wait_asynccnt(0);
#else
  asm volatile("s_wait_asynccnt 0x0" ::: "memory");
#endif
}

// ---------------- helpers ----------------
__device__ __forceinline__ float sigm1024(float x) {
  return 1024.f / (1.f + __builtin_expf(-x));
}

// ---------------- kernel 1: points + dedup + gaussian constants ----------------
__global__ void k_points(const float* __restrict__ Dp_, const float* __restrict__ sigma_,
                         const float* __restrict__ sel, const float* __restrict__ glob,
                         int* __restrict__ px, int* __restrict__ py, float* __restrict__ keep,
                         unsigned* __restrict__ claim,
                         float* __restrict__ gX, float* __restrict__ gY,
                         float* __restrict__ gInv, float* __restrict__ gCoef) {
  int p = blockIdx.x * blockDim.x + threadIdx.x;
  if (p < N_GAUSS) {
    float dx = sigm1024(Dp_[2 * p]);
    float dy = sigm1024(Dp_[2 * p + 1]);
    float sp = sigma_[p] + SIGMA_BOOST;
    float splus = fmaxf(sp, 0.f) + log1pf(__builtin_expf(-fabsf(sp)));
    float var = splus * (float)INPUT_SIZE * 0.1f + TAU; // variance per dim
    gX[p] = dx;
    gY[p] = dy;
    gInv[p] = 1.f / var;
    gCoef[p] = 1.f / (2.f * 3.14159265358979323846f * var); // 1/(2*pi*sigma^2)
  }
  if (p >= NPTS) return;

  float X, Y;
  if (p < P_NEAR) {
    int g = p >> 2, c = p & 3; // corners [[0,0],[0,1],[1,0],[1,1]]
    X = sigm1024(Dp_[2 * g]) + (float)(c >> 1);
    Y = sigm1024(Dp_[2 * g + 1]) + (float)(c & 1);
  } else if (p < P_NEAR + P_LOC) {
    int i = p - P_NEAR;
    int g = i >> 3;
    X = sigm1024(Dp_[2 * g]) + (sel[2 * i] - 0.5f) * 10.f; // log2(1024)=10
    Y = sigm1024(Dp_[2 * g + 1]) + (sel[2 * i + 1] - 0.5f) * 10.f;
  } else {
    int i = p - (P_NEAR + P_LOC);
    X = glob[2 * i] * (float)INPUT_SIZE;
    Y = glob[2 * i + 1] * (float)OUTPUT_SIZE;
  }
  float rx = __builtin_rintf(X), ry = __builtin_rintf(Y); // round-half-even, then clip
  int ix = (int)fminf(fmaxf(rx, 0.f), (float)(INPUT_SIZE - 1));
  int iy = (int)fminf(fmaxf(ry, 0.f), (float)(OUTPUT_SIZE - 1));
  px[p] = ix;
  py[p] = iy;
  // Dedup: first claimer of a grid cell keeps it. Duplicate points have
  // identical prob rows, so the winner's identity cannot change the output.
  unsigned cell = (unsigned)ix * OUTPUT_SIZE + (unsigned)iy;
  unsigned old = atomicOr(&claim[cell >> 5], 1u << (cell & 31u));
  keep[p] = ((old >> (cell & 31u)) & 1u) ? 0.f : 1.f;
}

// ---------------- kernel 2: colsum[g] = sum_p keep*prob, vc[g] = v/colsum ----------------
__global__ void k_colsum(const int* __restrict__ px, const int* __restrict__ py,
                         const float* __restrict__ keep,
                         const float* __restrict__ gX, const float* __restrict__ gY,
                         const float* __restrict__ gInv, const float* __restrict__ gCoef,
                         const float* __restrict__ v, float* __restrict__ colsum,
                         float* __restrict__ vc) {
  int g = blockIdx.x;
  float Dx = gX[g], Dy = gY[g], inv = gInv[g], coef = gCoef[g];
  float acc = 0.f;
  for (int p = threadIdx.x; p < NPTS; p += blockDim.x) {
    float dx = (float)px[p] - Dx;
    float dy = (float)py[p] - Dy;
    acc += keep[p] * coef * __builtin_expf(-0.5f * inv * (dx * dx + dy * dy));
  }
  __shared__ float red[256];
  red[threadIdx.x] = acc;
  __syncthreads();
  for (int s = 128; s > 0; s >>= 1) {
    if ((int)threadIdx.x < s) red[threadIdx.x] += red[threadIdx.x + s];
    __syncthreads();
  }
  if (threadIdx.x == 0) {
    float cs = red[0];
    colsum[g] = cs;
    vc[g] = v[g] / cs;
  }
}

// ---------------- kernel 3: v'[p] = keep * sum_g prob * v/colsum ----------------
__global__ void k_vprime(const int* __restrict__ px, const int* __restrict__ py,
                         const float* __restrict__ keep,
                         const float* __restrict__ gX, const float* __restrict__ gY,
                         const float* __restrict__ gInv, const float* __restrict__ gCoef,
                         const float* __restrict__ vc, float* __restrict__ vp) {
  int p = blockIdx.x;
  float X = (float)px[p], Y = (float)py[p];
  float acc = 0.f;
  for (int g = threadIdx.x; g < N_GAUSS; g += blockDim.x) {
    float dx = X - gX[g];
    float dy = Y - gY[g];
    acc += gCoef[g] * __builtin_expf(-0.5f * gInv[g] * (dx * dx + dy * dy)) * vc[g];
  }
  __shared__ float red[256];
  red[threadIdx.x] = acc;
  __syncthreads();
  for (int s = 128; s > 0; s >>= 1) {
    if ((int)threadIdx.x < s) red[threadIdx.x] += red[threadIdx.x + s];
    __syncthreads();
  }
  if (threadIdx.x == 0) vp[p] = keep[p] * red[0];
}

// ---------------- kernel 4: densify W[row,col] += v' ----------------
__global__ void k_scatter(const int* __restrict__ px, const int* __restrict__ py,
                          const float* __restrict__ vp, float* __restrict__ W) {
  int p = blockIdx.x * blockDim.x + threadIdx.x;
  if (p >= NPTS) return;
  float val = vp[p];
  if (val != 0.f) atomicAdd(&W[(size_t)px[p] * OUTPUT_SIZE + py[p]], val);
}

// ---------------- kernel 5: out(4096x1024) = x(4096x1024) @ W(1024x1024) ----------------
// fp32 WMMA 16x16x4. Block tile 128(M)x64(N), 8 waves in 4x2; each wave 32x32
// (4 accumulators) so every A/B fragment feeds 2 WMMAs. K-chunks of 32 are
// double-buffered in LDS via CDNA5 async global->LDS copies (ASYNCcnt).
#define BM 128
#define BN 64
#define BK 32
#define LDA 36 /* padded row stride (floats); 144 B: 16B aligned, conflict-free */
#define LDB 68 /* padded row stride (floats); 272 B: 16B aligned, conflict-free */

__launch_bounds__(256) __global__
void k_gemm(const float* __restrict__ x, const float* __restrict__ W,
            float* __restrict__ out) {
  __shared__ float As[2][BM * LDA]; // 2 x 18432 B
  __shared__ float Bs[2][BK * LDB]; // 2 x 8704 B

  const int tid = threadIdx.x;
  const int wave = tid >> 5;
  const int lane = tid & 31;
  const int hf = lane >> 4; // lane half: K-pair (A/B frags), M-half (C/D)
  const int lo = lane & 15;
  const int wm = wave & 3;  // 4 waves along M (32 rows each)
  const int wn = wave >> 2; // 2 waves along N (32 cols each)
  const int m0 = blockIdx.x * BM;
  const int n0 = blockIdx.y * BN;

  v8f acc00 = {}, acc01 = {}, acc10 = {}, acc11 = {};

  auto stageA = [&](int buf, int kc) {
#pragma unroll
    for (int it = 0; it < 4; ++it) { // 128x32 floats = 1024 x 16B chunks
      int idx = it * 256 + tid;
      int r = idx >> 3, c4 = (idx & 7) << 2;
      const float* g = &x[(size_t)(m0 + r) * INPUT_SIZE + kc + c4];
      float* l = &As[buf][r * LDA + c4];
#ifdef HAVE_ASYNC_LDS
      __builtin_amdgcn_global_load_async_to_lds_b128((GV4i*)(float*)g, (LV4i*)l, 0, 0);
#else
      *(v4f*)l = *(const v4f*)g;
#endif
    }
  };
  auto stageB = [&](int buf, int kc) {
#pragma unroll
    for (int it = 0; it < 2; ++it) { // 32x64 floats = 512 x 16B chunks
      int idx = it * 256 + tid;
      int r = idx >> 4, c4 = (idx & 15) << 2;
      const float* g = &W[(size_t)(kc + r) * OUTPUT_SIZE + n0 + c4];
      float* l = &Bs[buf][r * LDB + c4];
#ifdef HAVE_ASYNC_LDS
      __builtin_amdgcn_global_load_async_to_lds_b128((GV4i*)(float*)g, (LV4i*)l, 0, 0);
#else
      *(v4f*)l = *(const v4f*)g;
#endif
    }
  };

  // prologue: fill buffer 0
  stageA(0, 0);
  stageB(0, 0);
#ifdef HAVE_ASYNC_LDS
  wait_async_zero();
#endif
  __syncthreads();

  int cur = 0;
  for (int kc = 0; kc < INPUT_SIZE; kc += BK) {
    const int nxt = cur ^ 1;
    const bool more = (kc + BK) < INPUT_SIZE;
    if (more) { // prefetch next chunk into the other buffer while computing
      stageA(nxt, kc + BK);
      stageB(nxt, kc + BK);
    }

    const float* Ab = As[cur];
    const float* Bb = Bs[cur];
#pragma unroll
    for (int ks = 0; ks < BK; ks += 4) {
      // A 16x4 frags: lanes 0-15 hold K=ks,ks+1; lanes 16-31 hold K=ks+2,ks+3
      v2f a0 = *(const v2f*)&Ab[(wm * 32 + lo) * LDA + ks + 2 * hf];
      v2f a1 = *(const v2f*)&Ab[(wm * 32 + 16 + lo) * LDA + ks + 2 * hf];
      int kb = ks + 2 * hf;
      // B 4x16 frags (mirror of A layout with N in place of M)
      v2f b0 = {Bb[kb * LDB + wn * 32 + lo], Bb[(kb + 1) * LDB + wn * 32 + lo]};
      v2f b1 = {Bb[kb * LDB + wn * 32 + 16 + lo], Bb[(kb + 1) * LDB + wn * 32 + 16 + lo]};
      acc00 = __builtin_amdgcn_wmma_f32_16x16x4_f32(false, a0, false, b0, (short)0, acc00,
                                                    false, false);
      acc01 = __builtin_amdgcn_wmma_f32_16x16x4_f32(false, a0, false, b1, (short)0, acc01,
                                                    false, false);
      acc10 = __builtin_amdgcn_wmma_f32_16x16x4_f32(false, a1, false, b0, (short)0, acc10,
                                                    false, false);
      acc11 = __builtin_amdgcn_wmma_f32_16x16x4_f32(false, a1, false, b1, (short)0, acc11,
                                                    false, false);
    }

#ifdef HAVE_ASYNC_LDS
    if (more) wait_async_zero(); // our copies into nxt have landed in LDS
#endif
    __syncthreads(); // everyone done reading cur + everyone's nxt copies visible
    cur = nxt;
  }

  // C/D layout: VGPR r -> rows r (lanes 0-15) and r+8 (lanes 16-31)
  const int ocol = n0 + wn * 32 + lo;
#pragma unroll
  for (int r = 0; r < 8; ++r) {
    int row0 = m0 + wm * 32 + hf * 8 + r;
    out[(size_t)row0 * OUTPUT_SIZE + ocol] = acc00[r];
    out[(size_t)row0 * OUTPUT_SIZE + ocol + 16] = acc01[r];
    int row1 = row0 + 16;
    out[(size_t)row1 * OUTPUT_SIZE + ocol] = acc10[r];
    out[(size_t)row1 * OUTPUT_SIZE + ocol + 16] = acc11[r];
  }
}

// ---------------- launcher ----------------
extern "C" void kernel_launch(void* const* d_in, const int* in_sizes, int n_in,
                              void* d_out, int out_size, void* d_ws, size_t ws_size,
                              hipStream_t stream) {
  (void)in_sizes;
  (void)n_in;
  (void)out_size;
  (void)ws_size;
  const float* x = (const float*)d_in[0];
  const float* Dp_ = (const float*)d_in[1];
  const float* sigma_ = (const float*)d_in[2];
  const float* v = (const float*)d_in[3];
  const float* sel = (const float*)d_in[4];
  const float* glob = (const float*)d_in[5];
  float* out = (float*)d_out;

  char* ws = (char*)d_ws;
  size_t off = 0;
  auto alloc = [&](size_t bytes) -> void* {
    void* p = ws + off;
    off = (off + bytes + 255) & ~(size_t)255;
    return p;
  };
  unsigned* claim = (unsigned*)alloc(32768 * sizeof(unsigned)); // 1M-cell bitmap
  int* px = (int*)alloc(NPTS * sizeof(int));
  int* py = (int*)alloc(NPTS * sizeof(int));
  float* keep = (float*)alloc(NPTS * sizeof(float));
  float* gX = (float*)alloc(N_GAUSS * sizeof(float));
  float* gY = (float*)alloc(N_GAUSS * sizeof(float));
  float* gInv = (float*)alloc(N_GAUSS * sizeof(float));
  float* gCoef = (float*)alloc(N_GAUSS * sizeof(float));
  float* colsum = (float*)alloc(N_GAUSS * sizeof(float));
  float* vc = (float*)alloc(N_GAUSS * sizeof(float));
  float* vp = (float*)alloc(NPTS * sizeof(float));
  float* W = (float*)alloc((size_t)INPUT_SIZE * OUTPUT_SIZE * sizeof(float));

  (void)hipMemsetAsync(claim, 0, 32768 * sizeof(unsigned), stream);
  (void)hipMemsetAsync(W, 0, (size_t)INPUT_SIZE * OUTPUT_SIZE * sizeof(float), stream);

  k_points<<<(NPTS + 255) / 256, 256, 0, stream>>>(Dp_, sigma_, sel, glob, px, py, keep,
                                                   claim, gX, gY, gInv, gCoef);
  k_colsum<<<N_GAUSS, 256, 0, stream>>>(px, py, keep, gX, gY, gInv, gCoef, v, colsum, vc);
  k_vprime<<<NPTS, 256, 0, stream>>>(px, py, keep, gX, gY, gInv, gCoef, vc, vp);
  k_scatter<<<(NPTS + 255) / 256, 256, 0, stream>>>(px, py, vp, W);

  dim3 grid(BATCH / BM, OUTPUT_SIZE / BN);
  k_gemm<<<grid, 256, 0, stream>>>(x, W, out);
}